// SmoothOrdinalLoss_69655779607155
// MI455X (gfx1250) — compile-verified
//
#include <hip/hip_runtime.h>
#include <stdint.h>

// SmoothOrdinalLoss for MI455X (gfx1250, wave32).
// Bandwidth-bound: 256 MB of logits streamed exactly once through the
// gfx1250 async global->LDS engine (global_load_async_to_lds_b128,
// ASYNCcnt double-buffer), per-row base-2 log-sum-exp + 17-tap soft-target
// dot (exact fp32 support of exp(-d^4/50) is |d|<=8), one atomic per block.

#define CLASSES 256
#define WPB 8                 // waves per block (wave32 -> 256 threads)
#define THREADS (WPB * 32)
#define LOG2E 1.4426950408889634f
#define LN2   0.6931471805599453f

__device__ __forceinline__ float fexp2(float x) { return __builtin_amdgcn_exp2f(x); }
__device__ __forceinline__ float flog2(float x) { return __builtin_amdgcn_logf(x); }

__device__ __forceinline__ float wred_add(float v) {
#pragma unroll
  for (int m = 16; m >= 1; m >>= 1) v += __shfl_xor(v, m, 32);
  return v;
}
__device__ __forceinline__ float wred_max(float v) {
#pragma unroll
  for (int m = 16; m >= 1; m >>= 1) v = fmaxf(v, __shfl_xor(v, m, 32));
  return v;
}

// Low 32 bits of a generic pointer into LDS == LDS byte offset
// (ISA 10.2: LDS aperture maps flat addr by truncation to addr[31:0]).
__device__ __forceinline__ uint32_t lds_lo32(const void* p) {
  return (uint32_t)(uintptr_t)p;
}

// Stage one 256-float row into LDS: 2 x (32 lanes * 16B) async transfers.
__device__ __forceinline__ void async_row_load(uint32_t lds_base,
                                               const float* __restrict__ grow,
                                               int lane) {
  uint32_t l0 = lds_base + (uint32_t)lane * 16u;
  const float* g0 = grow + lane * 4;
  asm volatile("global_load_async_to_lds_b128 %0, %1, off"
               :: "v"(l0), "v"(g0) : "memory");
  uint32_t l1 = l0 + 512u;
  const float* g1 = g0 + 128;
  asm volatile("global_load_async_to_lds_b128 %0, %1, off"
               :: "v"(l1), "v"(g1) : "memory");
}

__global__ void __launch_bounds__(THREADS)
smooth_ordinal_loss_kernel(const float* __restrict__ logits,
                           const int*   __restrict__ targets,
                           float*       __restrict__ out,
                           int batch, int chunk, float inv_batch) {
  __shared__ __align__(16) float tile[WPB][2][CLASSES];  // 16 KB double buffer
  __shared__ float red[WPB];

  const int lane = threadIdx.x & 31;
  const int w    = threadIdx.x >> 5;

  const int gw = (int)blockIdx.x * WPB + w;
  const int r0 = gw * chunk;
  const int r1 = (r0 + chunk < batch) ? (r0 + chunk) : batch;

  const uint32_t lds0 = lds_lo32(&tile[w][0][0]);
  const uint32_t lds1 = lds_lo32(&tile[w][1][0]);

  // ---- per-wave loop invariants: soft-target weights around the target ----
  // lane j in [0,16] owns offset d = j-8;  w = exp(-d^4/50), ln w = -d^4/50.
  float wv = 0.0f, wvlw = 0.0f, lwv = 0.0f;
  if (lane < 17) {
    const float d  = (float)(lane - 8);
    const float d2 = d * d;
    lwv  = -0.02f * d2 * d2;
    wv   = fexp2(lwv * LOG2E);
    wvlw = wv * lwv;
  }
  const float A0    = wred_add(wv);      // unclipped normalizer
  const float B0    = wred_add(wvlw);    // unclipped sum w*ln(w)
  const float invA0 = 1.0f / A0;
  const float C0    = B0 * invA0 - LN2 * flog2(A0);  // neg-entropy constant

  float acc = 0.0f;

  if (r0 < r1) {
    async_row_load(lds0, logits + (size_t)r0 * CLASSES, lane);
    int buf = 0;
    for (int r = r0; r < r1; ++r) {
      const bool has_next = (r + 1 < r1);
      if (has_next)
        async_row_load(buf ? lds0 : lds1, logits + (size_t)(r + 1) * CLASSES, lane);

      // In-order ASYNCcnt: waiting <=2 guarantees the current row's pair
      // landed while keeping the prefetched pair in flight.
      if (has_next) asm volatile("s_wait_asynccnt 0x2" ::: "memory");
      else          asm volatile("s_wait_asynccnt 0x0" ::: "memory");

      const float* rowp = &tile[w][buf][0];
      const float4* t4 = reinterpret_cast<const float4*>(rowp);
      float4 x0 = t4[lane];       // classes [4L .. 4L+3]
      float4 x1 = t4[32 + lane];  // classes [128+4L .. 128+4L+3]

      // --- log-sum-exp over 256 classes, base-2 domain (v_exp_f32/v_log_f32)
      float m = fmaxf(fmaxf(fmaxf(x0.x, x0.y), fmaxf(x0.z, x0.w)),
                      fmaxf(fmaxf(x1.x, x1.y), fmaxf(x1.z, x1.w)));
      m = wred_max(m);
      const float m2 = m * LOG2E;
      float s;
      s  = fexp2(fmaf(x0.x, LOG2E, -m2));
      s += fexp2(fmaf(x0.y, LOG2E, -m2));
      s += fexp2(fmaf(x0.z, LOG2E, -m2));
      s += fexp2(fmaf(x0.w, LOG2E, -m2));
      s += fexp2(fmaf(x1.x, LOG2E, -m2));
      s += fexp2(fmaf(x1.y, LOG2E, -m2));
      s += fexp2(fmaf(x1.z, LOG2E, -m2));
      s += fexp2(fmaf(x1.w, LOG2E, -m2));
      s = wred_add(s);                          // sum exp(x - m)
      const float lse = fmaf(LN2, flog2(s), m); // m + ln(sum)

      const int t = targets[r];  // wave-uniform
      if (t >= 8 && t <= CLASSES - 9) {
        // Fast path (no window clipping): only the dot needs reducing.
        float dwl = 0.0f;
        if (lane < 17) dwl = wv * rowp[t - 8 + lane];
        const float dwt = wred_add(dwl);
        acc += fmaf(-dwt, invA0, C0) + lse;
      } else {
        // Boundary rows: window clipped at [0, C), recompute A, Sum w*ln w.
        float aw = 0.0f, bw = 0.0f, dw = 0.0f;
        if (lane < 17) {
          const int c = t - 8 + lane;
          if (c >= 0 && c < CLASSES) {
            aw = wv; bw = wvlw; dw = wv * rowp[c];
          }
        }
        aw = wred_add(aw);
        bw = wred_add(bw);
        dw = wred_add(dw);
        acc += (bw - dw) / aw - LN2 * flog2(aw) + lse;
      }
      buf ^= 1;
    }
  }

  if (lane == 0) red[w] = acc;
  __syncthreads();
  if (w == 0) {
    float v = (lane < WPB) ? red[lane] : 0.0f;
    v = wred_add(v);
    if (lane == 0) atomicAdd(out, v * inv_batch);
  }
}

__global__ void zero_out_kernel(float* out) { out[0] = 0.0f; }

extern "C" void kernel_launch(void* const* d_in, const int* in_sizes, int n_in,
                              void* d_out, int out_size, void* d_ws, size_t ws_size,
                              hipStream_t stream) {
  const float* logits  = (const float*)d_in[0];
  const int*   targets = (const int*)d_in[1];
  float*       out     = (float*)d_out;

  const int batch = in_sizes[1];                 // targets count == rows
  const float inv_b = 1.0f / (float)batch;

  zero_out_kernel<<<1, 1, 0, stream>>>(out);     // d_out is poisoned; atomics follow

  // ~32 rows per wave -> deep async pipeline; 1024 blocks at B=262144.
  int blocks = (batch + (WPB * 32) - 1) / (WPB * 32);
  if (blocks < 1) blocks = 1;
  if (blocks > 4096) blocks = 4096;

  const int total_waves = blocks * WPB;
  const int chunk = (batch + total_waves - 1) / total_waves;  // host-side: no device idiv

  smooth_ordinal_loss_kernel<<<blocks, THREADS, 0, stream>>>(
      logits, targets, out, batch, chunk, inv_b);
}